// Spatial_AttLayer_1623497638565
// MI455X (gfx1250) — compile-verified
//
#include <hip/hip_runtime.h>
#include <hip/hip_bf16.h>
#include <math.h>

// Problem constants
#define NB   64          // batch
#define CC   64          // channels
#define TT   128         // frames
#define VV   25          // joints
#define HH   3           // heads
#define DQ   32
#define OO   (2*HH*DQ)   // 192 qkv output channels
#define PP   (TT*VV)     // 3200 positions per (n, channel)
#define XSTR (CC*PP)     // 204800  x batch stride
#define QSTR (OO*PP)     // 614400  qk / y batch stride
#define KCONV (OO*5)     // 960 conv im2col depth

typedef __attribute__((ext_vector_type(2))) float v2f;
typedef __attribute__((ext_vector_type(8))) float v8f;

__device__ __forceinline__ v8f wmma4(v2f a, v2f b, v8f c) {
  // D = A(16x4 f32) * B(4x16 f32) + C(16x16 f32)
  return __builtin_amdgcn_wmma_f32_16x16x4_f32(false, a, false, b, (short)0, c,
                                               false, false);
}

// ---------------------------------------------------------------------------
// K1: qk[n,o,p] = sum_c (x[n,c,p] + pe[c, p%25]) * Wqkv[o,c] + bqkv[o]
// one wave per 16(o) x 16(p) tile; K = 64
// ---------------------------------------------------------------------------
__global__ __launch_bounds__(256) void k_qkv(const float* __restrict__ x,
                                             const float* __restrict__ W,
                                             const float* __restrict__ bias,
                                             float* __restrict__ qk) {
  __shared__ float pe[CC * VV];
  for (int i = threadIdx.x; i < CC * VV; i += 256) {
    int c = i / VV, v = i % VV;
    int j = c >> 1;
    float div = expf(-(float)(2 * j) * 0.14391156516029343f); // ln(1e4)/64
    float ang = (float)v * div;
    pe[i] = (c & 1) ? cosf(ang) : sinf(ang);
  }
  __syncthreads();

  int wave = threadIdx.x >> 5;
  int lane = threadIdx.x & 31;
  int half = lane >> 4;
  int l    = lane & 15;

  int tile = blockIdx.x * 8 + wave;          // 153600 tiles total (exact grid)
  int n    = tile / 2400;
  int rem  = tile % 2400;
  int o0   = (rem / 200) * 16;
  int p0   = (rem % 200) * 16;

  const float* xn = x + (size_t)n * XSTR;
  int p = p0 + l;
  int vcol = p % VV;

  v8f acc = {};
  #pragma unroll
  for (int k0 = 0; k0 < CC; k0 += 4) {
    int ka = k0 + 2 * half;
    v2f a, b;
    a.x = W[(o0 + l) * CC + ka];
    a.y = W[(o0 + l) * CC + ka + 1];
    b.x = xn[ka * PP + p]       + pe[ka * VV + vcol];
    b.y = xn[(ka + 1) * PP + p] + pe[(ka + 1) * VV + vcol];
    acc = wmma4(a, b, acc);
  }

  float* qn = qk + (size_t)n * QSTR;
  #pragma unroll
  for (int i = 0; i < 8; i++) {
    int o = o0 + i + half * 8;
    qn[o * PP + p0 + l] = acc[i] + bias[o];
  }
}

// ---------------------------------------------------------------------------
// K2: att[n,h,u,v] = tanh( (sum_{c,t} q[...u] * k[...v]) / (DQ*T) )*alpha + att0
// one wave per 16x16 tile of the (25->32)x(25->32) score matrix; K = 4096
// ---------------------------------------------------------------------------
__global__ __launch_bounds__(256) void k_att(const float* __restrict__ qk,
                                             const float* __restrict__ alphas,
                                             const float* __restrict__ att0,
                                             float* __restrict__ att) {
  int wave = threadIdx.x >> 5;
  int lane = threadIdx.x & 31;
  int half = lane >> 4;
  int l    = lane & 15;

  int tile = blockIdx.x * 8 + wave;          // 768 tiles (exact grid)
  int n    = tile / 12;
  int rem  = tile % 12;
  int h    = rem / 4;
  int q4   = rem % 4;
  int u0   = (q4 >> 1) * 16;
  int v0   = (q4 & 1) * 16;

  const float* qb = qk + (size_t)n * QSTR + (size_t)(h * DQ) * PP;
  const float* kb = qk + (size_t)n * QSTR + (size_t)(HH * DQ + h * DQ) * PP;

  int u = u0 + l;  bool uok = u < VV;
  int v = v0 + l;  bool vok = v < VV;

  v8f acc = {};
  for (int c = 0; c < DQ; c++) {
    const float* qc = qb + c * PP;
    const float* kc = kb + c * PP;
    #pragma unroll 8
    for (int t = 0; t < TT; t += 4) {
      int ka = t + 2 * half;
      v2f a, b;
      a.x = uok ? qc[ka * VV + u]       : 0.0f;
      a.y = uok ? qc[(ka + 1) * VV + u] : 0.0f;
      b.x = vok ? kc[ka * VV + v]       : 0.0f;
      b.y = vok ? kc[(ka + 1) * VV + v] : 0.0f;
      acc = wmma4(a, b, acc);
    }
  }

  float alpha = alphas[h];
  const float* a0 = att0 + h * VV * VV;
  float* S = att + (size_t)n * (HH * VV * VV) + h * VV * VV;
  const float inv = 1.0f / (float)(DQ * TT);
  #pragma unroll
  for (int i = 0; i < 8; i++) {
    int uu = u0 + i + half * 8;
    int vv = v0 + l;
    if (uu < VV && vv < VV)
      S[uu * VV + vv] = tanhf(acc[i] * inv) * alpha + a0[uu * VV + vv];
  }
}

// ---------------------------------------------------------------------------
// K3: y[n, h*64+c, t, v] = sum_u x[n,c,t,u] * att[n,h,u,v]
// rows r = c*T + t (8192 per (n,h)); one wave per 16(r) x 16(v) tile; K = 25
// ---------------------------------------------------------------------------
__global__ __launch_bounds__(256) void k_yatt(const float* __restrict__ x,
                                              const float* __restrict__ att,
                                              float* __restrict__ y) {
  int wave = threadIdx.x >> 5;
  int lane = threadIdx.x & 31;
  int half = lane >> 4;
  int l    = lane & 15;

  int tile = blockIdx.x * 8 + wave;          // 196608 tiles (exact grid)
  int n    = tile / 3072;
  int rem  = tile % 3072;
  int h    = rem / 1024;
  int rem2 = rem % 1024;
  int r0   = (rem2 >> 1) * 16;
  int v0   = (rem2 & 1) * 16;

  const float* xn = x + (size_t)n * XSTR;
  const float* A  = att + (size_t)n * (HH * VV * VV) + h * VV * VV;

  int v = v0 + l;  bool vok = v < VV;
  int r = r0 + l;

  v8f acc = {};
  #pragma unroll
  for (int k0 = 0; k0 < 28; k0 += 4) {       // K = 25, padded to 28
    int ka = k0 + 2 * half;
    v2f a, b;
    a.x = (ka < VV)            ? xn[r * VV + ka]     : 0.0f;
    a.y = (ka + 1 < VV)        ? xn[r * VV + ka + 1] : 0.0f;
    b.x = (ka < VV && vok)     ? A[ka * VV + v]       : 0.0f;
    b.y = (ka + 1 < VV && vok) ? A[(ka + 1) * VV + v] : 0.0f;
    acc = wmma4(a, b, acc);
  }

  float* yn = y + (size_t)n * QSTR + (size_t)h * XSTR;
  #pragma unroll
  for (int i = 0; i < 8; i++) {
    int rr = r0 + i + half * 8;
    int vv = v0 + l;
    if (vv < VV) yn[rr * VV + vv] = acc[i];
  }
}

// ---------------------------------------------------------------------------
// K4: conv 1x5 over V: out[n,co,t,v] = bout[co] +
//        sum_{ci,dv} Wout[co,ci,0,dv] * y[n,ci,t,v+dv-2]
// im2col GEMM: M=64(co), K=960, N=(t,v); one wave per 16x16 tile
// ---------------------------------------------------------------------------
__global__ __launch_bounds__(256) void k_conv(const float* __restrict__ y,
                                              const float* __restrict__ Wout,
                                              const float* __restrict__ bout,
                                              float* __restrict__ out) {
  int wave = threadIdx.x >> 5;
  int lane = threadIdx.x & 31;
  int half = lane >> 4;
  int l    = lane & 15;

  int tile = blockIdx.x * 8 + wave;          // 51200 tiles (exact grid)
  int n    = tile / 800;
  int rem  = tile % 800;
  int p0   = (rem / 4) * 16;
  int co0  = (rem % 4) * 16;

  const float* yn = y + (size_t)n * QSTR;
  int p = p0 + l;
  int t = p / VV;
  int v = p % VV;

  v8f acc = {};
  for (int k0 = 0; k0 < KCONV; k0 += 4) {
    int ka = k0 + 2 * half;
    v2f a, b;
    a.x = Wout[(co0 + l) * KCONV + ka];
    a.y = Wout[(co0 + l) * KCONV + ka + 1];
    int ci0 = ka / 5, dv0 = ka % 5, vs0 = v + dv0 - 2;
    int ci1 = (ka + 1) / 5, dv1 = (ka + 1) % 5, vs1 = v + dv1 - 2;
    b.x = (vs0 >= 0 && vs0 < VV) ? yn[ci0 * PP + t * VV + vs0] : 0.0f;
    b.y = (vs1 >= 0 && vs1 < VV) ? yn[ci1 * PP + t * VV + vs1] : 0.0f;
    acc = wmma4(a, b, acc);
  }

  float* on = out + (size_t)n * XSTR;
  #pragma unroll
  for (int i = 0; i < 8; i++) {
    int co = co0 + i + half * 8;
    on[co * PP + p0 + l] = acc[i] + bout[co];
  }
}

// ---------------------------------------------------------------------------
// K5: per-channel batch mean/var over (N,T,V); one block per channel
// ---------------------------------------------------------------------------
__global__ __launch_bounds__(256) void k_stats(const float* __restrict__ yc,
                                               float* __restrict__ stats) {
  int c = blockIdx.x;
  double s = 0.0, s2 = 0.0;
  for (int j = threadIdx.x; j < NB * PP; j += 256) {
    int n = j / PP, r = j % PP;
    float val = yc[(size_t)n * XSTR + c * PP + r];
    s  += (double)val;
    s2 += (double)val * (double)val;
  }
  __shared__ double sh[256], sh2[256];
  sh[threadIdx.x] = s;
  sh2[threadIdx.x] = s2;
  __syncthreads();
  for (int st = 128; st > 0; st >>= 1) {
    if (threadIdx.x < st) {
      sh[threadIdx.x]  += sh[threadIdx.x + st];
      sh2[threadIdx.x] += sh2[threadIdx.x + st];
    }
    __syncthreads();
  }
  if (threadIdx.x == 0) {
    double cnt = (double)(NB * PP);
    double m   = sh[0] / cnt;
    double var = sh2[0] / cnt - m * m;
    stats[c]      = (float)m;
    stats[CC + c] = (float)var;
  }
}

// ---------------------------------------------------------------------------
// K6: out = leaky_relu(batchnorm(out) + x), in place on d_out
// ---------------------------------------------------------------------------
__global__ __launch_bounds__(256) void k_final(const float* __restrict__ x,
                                               const float* __restrict__ gamma,
                                               const float* __restrict__ beta,
                                               const float* __restrict__ stats,
                                               float* __restrict__ out) {
  size_t i = (size_t)blockIdx.x * 256 + threadIdx.x;   // exact: 13,107,200
  int c = (int)((i / PP) % CC);
  float m = stats[c], var = stats[CC + c];
  float val = (out[i] - m) * rsqrtf(var + 1e-5f) * gamma[c] + beta[c] + x[i];
  out[i] = val > 0.0f ? val : 0.1f * val;
}

// ---------------------------------------------------------------------------
extern "C" void kernel_launch(void* const* d_in, const int* in_sizes, int n_in,
                              void* d_out, int out_size, void* d_ws, size_t ws_size,
                              hipStream_t stream) {
  const float* x      = (const float*)d_in[0];
  const float* Wqkv   = (const float*)d_in[1];
  const float* bqkv   = (const float*)d_in[2];
  const float* alphas = (const float*)d_in[3];
  const float* att0s  = (const float*)d_in[4];
  const float* Wout   = (const float*)d_in[5];
  const float* bout   = (const float*)d_in[6];
  const float* gamma  = (const float*)d_in[7];
  const float* beta   = (const float*)d_in[8];
  float* out = (float*)d_out;

  // workspace layout: qk (reused as y) | att | stats
  float* qk    = (float*)d_ws;                              // 39,321,600 f32
  float* att   = qk + (size_t)NB * QSTR;                    // 120,000 f32
  float* stats = att + (size_t)NB * HH * VV * VV;           // 128 f32
  float* y     = qk;   // qk is dead after k_att; reuse the region for y

  k_qkv  <<<19200, 256, 0, stream>>>(x, Wqkv, bqkv, qk);
  k_att  <<<96,    256, 0, stream>>>(qk, alphas, att0s, att);
  k_yatt <<<24576, 256, 0, stream>>>(x, att, y);
  k_conv <<<6400,  256, 0, stream>>>(y, Wout, bout, out);
  k_stats<<<64,    256, 0, stream>>>(out, stats);
  k_final<<<51200, 256, 0, stream>>>(x, gamma, beta, stats, out);
}